// DynamicMultiMetricLoss_12352325944204
// MI455X (gfx1250) — compile-verified
//
#include <hip/hip_runtime.h>
#include <hip/hip_bf16.h>
#include <math.h>

// Problem constants (fixed by the reference setup)
#define EPSF   1e-8f
#define CCH    128
#define HWSZ   65536      // H*W
#define NPIX   262144     // B*H*W
#define NCHUNK 64
#define CHUNK  4096       // NPIX / NCHUNK
#define TOPK   13107      // int(262144 * 0.05)

typedef float v2f __attribute__((ext_vector_type(2)));
typedef float v8f __attribute__((ext_vector_type(8)));
typedef unsigned int uint32;

// ---------------------------------------------------------------------------
// Pass 0: per-channel partial sums (x, x^2, y, y^2) via V_WMMA_F32_16X16X4_F32.
// Each wave owns one (16-channel tile, 4096-pixel chunk). A = data [16ch x 4px]
// (lanes 0-15: K=0,1 ; lanes 16-31: K=2,3), B = all-ones -> D rows are channel
// sums, replicated over N. fp32 WMMA => exact same precision as scalar adds.
// ---------------------------------------------------------------------------
__global__ void bn_partial_wmma(const float* __restrict__ rec,
                                const float* __restrict__ tgt,
                                float* __restrict__ partials) {
    const int wave = threadIdx.x >> 5;
    const int lane = threadIdx.x & 31;
    const int gw   = blockIdx.x * 8 + wave;   // 0..511
    const int ct   = gw >> 6;                 // channel tile 0..7
    const int q    = gw & 63;                 // pixel chunk 0..63
    const int m    = lane & 15;
    const int kq   = lane >> 4;               // 0 or 1
    const int c    = ct * 16 + m;
    const int pstart = q * CHUNK;
    const int b    = pstart >> 16;            // / HWSZ
    const int hw0  = pstart & (HWSZ - 1);
    const size_t base = (size_t)(b * CCH + c) * HWSZ + hw0;

    v8f ax  = {0.f,0.f,0.f,0.f,0.f,0.f,0.f,0.f};
    v8f axx = ax, ay = ax, ayy = ax;
    const v2f ones = {1.0f, 1.0f};

    for (int k = 0; k < CHUNK / 4; ++k) {
        const size_t off = base + (size_t)(4 * k + 2 * kq);
        v2f xa = *(const v2f*)(rec + off);
        v2f ya = *(const v2f*)(tgt + off);
        v2f xs = xa * xa;
        v2f ys = ya * ya;
        ax  = __builtin_amdgcn_wmma_f32_16x16x4_f32(false, xa, false, ones, (short)0, ax,  false, false);
        axx = __builtin_amdgcn_wmma_f32_16x16x4_f32(false, xs, false, ones, (short)0, axx, false, false);
        ay  = __builtin_amdgcn_wmma_f32_16x16x4_f32(false, ya, false, ones, (short)0, ay,  false, false);
        ayy = __builtin_amdgcn_wmma_f32_16x16x4_f32(false, ys, false, ones, (short)0, ayy, false, false);
    }

    // C/D layout: lanes 0-15 VGPR r = D[r][lane] (channels ct*16+0..7),
    // lanes 16-31 VGPR r = D[r+8][lane-16] (channels ct*16+8..15).
    if (lane == 0 || lane == 16) {
        const int chbase = (lane == 0) ? 0 : 8;
#pragma unroll
        for (int r = 0; r < 8; ++r) {
            float* p = partials + (((size_t)(ct * 64 + q) * 16 + (chbase + r)) * 4);
            p[0] = ax[r]; p[1] = axx[r]; p[2] = ay[r]; p[3] = ayy[r];
        }
    }
}

// ---------------------------------------------------------------------------
// Pass 0b: reduce chunk partials -> per-channel mean and 1/(std+eps) (ddof=1),
// and softmax of the 9 loss weights.
// ---------------------------------------------------------------------------
__global__ void bn_finalize(const float* __restrict__ partials,
                            const float* __restrict__ w_in,
                            float4* __restrict__ bn_out,
                            float* __restrict__ w_out) {
    const int c = threadIdx.x;          // 0..127
    const int ct = c >> 4, chl = c & 15;
    float sx = 0.f, sxx = 0.f, sy = 0.f, syy = 0.f;
    for (int q = 0; q < 64; ++q) {
        const float* p = partials + (((size_t)(ct * 64 + q) * 16 + chl) * 4);
        sx += p[0]; sxx += p[1]; sy += p[2]; syy += p[3];
    }
    const float N = (float)NPIX;
    float mx = sx / N;
    float vx = fmaxf((sxx - sx * sx / N) / (N - 1.0f), 0.f);
    float my = sy / N;
    float vy = fmaxf((syy - sy * sy / N) / (N - 1.0f), 0.f);
    bn_out[c] = make_float4(mx, 1.0f / (sqrtf(vx) + EPSF),
                            my, 1.0f / (sqrtf(vy) + EPSF));
    if (c == 0) {
        float wv[9], wm = -3.4e38f, s = 0.f;
        for (int i = 0; i < 9; ++i) { wv[i] = w_in[i]; wm = fmaxf(wm, wv[i]); }
        for (int i = 0; i < 9; ++i) { wv[i] = __expf(wv[i] - wm); s += wv[i]; }
        for (int i = 0; i < 9; ++i) w_out[i] = wv[i] / s;
    }
}

// ---------------------------------------------------------------------------
// Helpers for closed-form SAM / Pearson from moments.
// ---------------------------------------------------------------------------
__device__ __forceinline__ void sam_pearson(float Sp, float St, float Spp, float Stt,
                                            float Spt, float& sam, float& pear) {
    float np = fmaxf(sqrtf(fmaxf(Spp, 0.f)), EPSF);
    float nt = fmaxf(sqrtf(fmaxf(Stt, 0.f)), EPSF);
    sam = 1.0f - Spt / (np * nt);
    const float invC = 1.0f / 128.0f;
    float cov = Spt - Sp * St * invC;
    float vp  = Spp - Sp * Sp * invC;
    float vt  = Stt - St * St * invC;
    pear = 1.0f - cov / (sqrtf(fmaxf(vp, 0.f) + EPSF) * sqrtf(fmaxf(vt, 0.f) + EPSF));
}

// p = A*x + Bc, t = Cc*y + Dc  ->  second-order sums from raw moments.
__device__ __forceinline__ void affine_moments(float A, float Bc, float Cc, float Dc,
                                               float Sx, float Sy, float Sxx, float Syy, float Sxy,
                                               float& Sp, float& St, float& Spp, float& Stt, float& Spt) {
    Sp  = A * Sx + 128.f * Bc;
    St  = Cc * Sy + 128.f * Dc;
    Spp = A * A * Sxx + 2.f * A * Bc * Sx + 128.f * Bc * Bc;
    Stt = Cc * Cc * Syy + 2.f * Cc * Dc * Sy + 128.f * Dc * Dc;
    Spt = A * Cc * Sxy + A * Dc * Sx + Bc * Cc * Sy + 128.f * Bc * Dc;
}

// ---------------------------------------------------------------------------
// Pass 1: one thread per pixel. Loop 1: raw + bn moments (coalesced, strided
// by HW over channels). Closed-form SAM/Pearson for all 3 norms. Loop 2 (L2-
// resident re-read): SID log sums, collapsed to sum (p-t)*(log p - log t).
// ---------------------------------------------------------------------------
__global__ void pixel_loss(const float* __restrict__ rec,
                           const float* __restrict__ tgt,
                           const float4* __restrict__ bn,
                           const float* __restrict__ wts,
                           float* __restrict__ out_map) {
    __shared__ float4 sBN[128];
    __shared__ float  sW[9];
    if (threadIdx.x < 128) sBN[threadIdx.x] = bn[threadIdx.x];
    if (threadIdx.x < 9)   sW[threadIdx.x]  = wts[threadIdx.x];
    __syncthreads();

    const int p  = blockIdx.x * blockDim.x + threadIdx.x;
    const int b  = p >> 16;
    const int hw = p & (HWSZ - 1);
    const size_t base = (size_t)b * CCH * HWSZ + hw;

    float sx = 0.f, sy = 0.f, sxx = 0.f, syy = 0.f, sxy = 0.f;
    float mnx = 3.4e38f, mxx = -3.4e38f, mny = 3.4e38f, mxy = -3.4e38f;
    float sbx = 0.f, sby = 0.f, sbxx = 0.f, sbyy = 0.f, sbxy = 0.f;
    float mbx = 3.4e38f, mby = 3.4e38f;

    for (int c = 0; c < 128; ++c) {
        const size_t o = base + (size_t)c * HWSZ;
        float x = rec[o];
        float y = tgt[o];
        __builtin_prefetch(rec + o + (size_t)8 * HWSZ, 0, 0);
        __builtin_prefetch(tgt + o + (size_t)8 * HWSZ, 0, 0);
        sx += x; sy += y;
        sxx = fmaf(x, x, sxx); syy = fmaf(y, y, syy); sxy = fmaf(x, y, sxy);
        mnx = fminf(mnx, x); mxx = fmaxf(mxx, x);
        mny = fminf(mny, y); mxy = fmaxf(mxy, y);
        float4 bp = sBN[c];
        float px = (x - bp.x) * bp.y;
        float py = (y - bp.z) * bp.w;
        sbx += px; sby += py;
        sbxx = fmaf(px, px, sbxx); sbyy = fmaf(py, py, sbyy); sbxy = fmaf(px, py, sbxy);
        mbx = fminf(mbx, px); mby = fminf(mby, py);
    }

    // --- norm parameters (min-max and z are per-pixel affine maps) ---
    float Amm_p = 1.0f / (mxx - mnx + EPSF), Bmm_p = -mnx * Amm_p;
    float Amm_t = 1.0f / (mxy - mny + EPSF), Bmm_t = -mny * Amm_t;
    float meanx = sx * (1.0f / 128.f);
    float varx  = fmaxf((sxx - sx * sx * (1.0f / 128.f)) * (1.0f / 127.f), 0.f);
    float Az_p  = 1.0f / (sqrtf(varx) + EPSF), Bz_p = -meanx * Az_p;
    float meany = sy * (1.0f / 128.f);
    float vary  = fmaxf((syy - sy * sy * (1.0f / 128.f)) * (1.0f / 127.f), 0.f);
    float Az_t  = 1.0f / (sqrtf(vary) + EPSF), Bz_t = -meany * Az_t;

    float Sp, St, Spp, Stt, Spt;
    float sam_mm, pear_mm, sam_z, pear_z, sam_bn, pear_bn;
    affine_moments(Amm_p, Bmm_p, Amm_t, Bmm_t, sx, sy, sxx, syy, sxy, Sp, St, Spp, Stt, Spt);
    sam_pearson(Sp, St, Spp, Stt, Spt, sam_mm, pear_mm);
    affine_moments(Az_p, Bz_p, Az_t, Bz_t, sx, sy, sxx, syy, sxy, Sp, St, Spp, Stt, Spt);
    sam_pearson(Sp, St, Spp, Stt, Spt, sam_z, pear_z);
    sam_pearson(sbx, sby, sbxx, sbyy, sbxy, sam_bn, pear_bn);

    // --- SID probability scales: p_pos = (x - min)*scale for mm and z ---
    float smm_p = Amm_p / ((sx - 128.f * mnx) * Amm_p + EPSF);
    float smm_t = Amm_t / ((sy - 128.f * mny) * Amm_t + EPSF);
    float sz_p  = Az_p  / ((sx - 128.f * mnx) * Az_p  + EPSF);
    float sz_t  = Az_t  / ((sy - 128.f * mny) * Az_t  + EPSF);
    float ibn_p = 1.0f / ((sbx - 128.f * mbx) + EPSF);
    float ibn_t = 1.0f / ((sby - 128.f * mby) + EPSF);

    // fold the 6 closed-form losses into the weighted result now; only the 3
    // SID accumulators remain live across loop 2.
    float wm = sW[0] * sam_mm + sW[1] * pear_mm
             + sW[3] * sam_z  + sW[4] * pear_z
             + sW[6] * sam_bn + sW[7] * pear_bn;

    float sid_mm = 0.f, sid_z = 0.f, sid_bn = 0.f;
    for (int c = 0; c < 128; ++c) {
        const size_t o = base + (size_t)c * HWSZ;
        float x = rec[o];
        float y = tgt[o];
        float dx = x - mnx, dy = y - mny;
        float p1 = fmaxf(dx * smm_p, EPSF), t1 = fmaxf(dy * smm_t, EPSF);
        sid_mm += (p1 - t1) * (__logf(p1) - __logf(t1));
        float p2 = fmaxf(dx * sz_p, EPSF), t2 = fmaxf(dy * sz_t, EPSF);
        sid_z  += (p2 - t2) * (__logf(p2) - __logf(t2));
        float4 bp = sBN[c];
        float px = (x - bp.x) * bp.y;
        float py = (y - bp.z) * bp.w;
        float p3 = fmaxf((px - mbx) * ibn_p, EPSF), t3 = fmaxf((py - mby) * ibn_t, EPSF);
        sid_bn += (p3 - t3) * (__logf(p3) - __logf(t3));
    }

    wm += sW[2] * sid_mm + sW[5] * sid_z + sW[8] * sid_bn;
    out_map[p] = wm;
}

// ---------------------------------------------------------------------------
// Pass 2: deterministic top-k mean via 3-level radix-histogram select
// (12 + 12 + 8 bits of the order-preserving uint mapping). Histograms use
// INTEGER atomics (order-independent => bit-deterministic). Each level is a
// full-grid pass over the 1 MiB map (L2 resident), instead of 32 serial
// single-block passes.
// ---------------------------------------------------------------------------
__device__ __forceinline__ uint32 f2ord(float f) {
    uint32 ub = __float_as_uint(f);
    return (ub & 0x80000000u) ? ~ub : (ub | 0x80000000u);
}

__global__ void topk_init(uint32* state) {
    state[0] = 0u;          // prefix (accumulates the kth-largest ordinal)
    ((int*)state)[1] = TOPK; // remaining rank within the filtered set
}

__global__ void topk_hist(const float* __restrict__ map, int n,
                          const uint32* __restrict__ state,
                          int shift, uint32 keepmask, int nbins,
                          int* __restrict__ hist) {
    const uint32 prefix = state[0] & keepmask;
    const int stride = gridDim.x * blockDim.x;
    for (int i = blockIdx.x * blockDim.x + threadIdx.x; i < n; i += stride) {
        uint32 u = f2ord(map[i]);
        if ((u & keepmask) == prefix)
            atomicAdd(&hist[(u >> shift) & (nbins - 1)], 1);
    }
}

__global__ void topk_scan(const int* __restrict__ hist, int nbins,
                          uint32* __restrict__ state, int shift) {
    __shared__ int part[256];
    const int tid = threadIdx.x;
    const int chunk = nbins >> 8;           // 16 (4096 bins) or 1 (256 bins)
    int s = 0;
    for (int j = 0; j < chunk; ++j)
        s += hist[nbins - 1 - (tid * chunk + j)];
    part[tid] = s;
    __syncthreads();
    if (tid == 0) {
        int krem = ((int*)state)[1];
        int cum = 0, t = 0;
        for (; t < 255; ++t) {
            if (cum + part[t] >= krem) break;
            cum += part[t];
        }
        int bin = nbins - 1 - t * chunk;    // fallback: top bin of chunk
        for (int j = 0; j < chunk; ++j) {
            int b = nbins - 1 - (t * chunk + j);
            int h = hist[b];
            if (cum + h >= krem) { bin = b; break; }
            cum += h;
        }
        state[0] |= ((uint32)bin) << shift;
        ((int*)state)[1] = krem - cum;      // rank remaining inside chosen bin
    }
}

__global__ void topk_partial(const float* __restrict__ map, int n,
                             const uint32* __restrict__ state,
                             float* __restrict__ fsum, int* __restrict__ fcnt) {
    __shared__ float ss[256];
    __shared__ int   sc[256];
    const uint32 kth = state[0];
    const int tid = threadIdx.x;
    float s = 0.f; int c = 0;
    const int stride = gridDim.x * blockDim.x;
    for (int i = blockIdx.x * blockDim.x + tid; i < n; i += stride) {
        float v = map[i];
        if (f2ord(v) > kth) { s += v; c++; }
    }
    ss[tid] = s; sc[tid] = c;
    __syncthreads();
    for (int h = 128; h > 0; h >>= 1) {
        if (tid < h) { ss[tid] += ss[tid + h]; sc[tid] += sc[tid + h]; }
        __syncthreads();
    }
    if (tid == 0) { fsum[blockIdx.x] = ss[0]; fcnt[blockIdx.x] = sc[0]; }
}

__global__ void topk_final(const float* __restrict__ fsum, const int* __restrict__ fcnt,
                           const uint32* __restrict__ state, float* __restrict__ out,
                           int nparts, int k) {
    float s = 0.f; int c = 0;
    for (int i = 0; i < nparts; ++i) { s += fsum[i]; c += fcnt[i]; }
    const uint32 kth = state[0];
    const uint32 ub = (kth & 0x80000000u) ? (kth & 0x7FFFFFFFu) : ~kth;
    const float kval = __uint_as_float(ub);
    out[0] = (s + (float)(k - c) * kval) / (float)k;
}

// ---------------------------------------------------------------------------
extern "C" void kernel_launch(void* const* d_in, const int* in_sizes, int n_in,
                              void* d_out, int out_size, void* d_ws, size_t ws_size,
                              hipStream_t stream) {
    const float* rec = (const float*)d_in[0];
    const float* tgt = (const float*)d_in[1];
    const float* lw  = (const float*)d_in[2];
    float* out = (float*)d_out;

    // Workspace layout (floats): all regions 16B aligned.
    float*  P        = (float*)d_ws;
    float*  partials = P;                          // [0, 32768)   128 KB
    float4* bn       = (float4*)(P + 32768);       // [32768, +512)
    float*  wts      = P + 32768 + 512;            // [33280, +16)
    int*    hist     = (int*)(P + 33296);          // [33296, +4096)
    uint32* state    = (uint32*)(P + 37392);       // [37392, +4)  {prefix, krem}
    float*  fsum     = P + 37396;                  // [37396, +64)
    int*    fcnt     = (int*)(P + 37460);          // [37460, +64)

    // main computation
    bn_partial_wmma<<<64, 256, 0, stream>>>(rec, tgt, partials);
    bn_finalize<<<1, 128, 0, stream>>>(partials, lw, bn, wts);
    pixel_loss<<<NPIX / 256, 256, 0, stream>>>(rec, tgt, bn, wts, out + 1);

    // deterministic top-k mean (radix-histogram select over f2ord ordinals)
    const float* map = out + 1;
    topk_init<<<1, 1, 0, stream>>>(state);

    hipMemsetAsync(hist, 0, 4096 * sizeof(int), stream);
    topk_hist<<<256, 256, 0, stream>>>(map, NPIX, state, 20, 0x00000000u, 4096, hist);
    topk_scan<<<1, 256, 0, stream>>>(hist, 4096, state, 20);

    hipMemsetAsync(hist, 0, 4096 * sizeof(int), stream);
    topk_hist<<<256, 256, 0, stream>>>(map, NPIX, state, 8, 0xFFF00000u, 4096, hist);
    topk_scan<<<1, 256, 0, stream>>>(hist, 4096, state, 8);

    hipMemsetAsync(hist, 0, 256 * sizeof(int), stream);
    topk_hist<<<256, 256, 0, stream>>>(map, NPIX, state, 0, 0xFFFFFF00u, 256, hist);
    topk_scan<<<1, 256, 0, stream>>>(hist, 256, state, 0);

    topk_partial<<<64, 256, 0, stream>>>(map, NPIX, state, fsum, fcnt);
    topk_final<<<1, 1, 0, stream>>>(fsum, fcnt, state, out, 64, TOPK);
}